// BaselineModel_15290083574239
// MI455X (gfx1250) — compile-verified
//
#include <hip/hip_runtime.h>

typedef __attribute__((ext_vector_type(2))) float v2f;
typedef __attribute__((ext_vector_type(8))) float v8f;

// ---------------- degree + norm ----------------
__global__ void deg_kernel(const int* __restrict__ dst, float* __restrict__ deg, int E) {
  int e = blockIdx.x * blockDim.x + threadIdx.x;
  if (e < E) unsafeAtomicAdd(&deg[dst[e]], 1.0f);
}

__global__ void rsqrt_kernel(float* __restrict__ deg, int N) {
  int i = blockIdx.x * blockDim.x + threadIdx.x;
  if (i < N) deg[i] = rsqrtf(deg[i] + 1.0f);
}

// h0[N,16] = concat(x[:, :8], x_mask[:, :8])
__global__ void build_h0(const float* __restrict__ x, const float* __restrict__ xm,
                         float* __restrict__ h0, int N) {
  int idx = blockIdx.x * blockDim.x + threadIdx.x;
  if (idx >= N * 16) return;
  int i = idx >> 4, j = idx & 15;
  h0[idx] = (j < 8) ? x[i * 10 + j] : xm[i * 10 + (j - 8)];
}

// ---------------- FP32 WMMA GEMM: Out[N,128] = A[N,K] @ W[K,128] ----------------
// Block: 256 threads = 8 waves; wave w computes the 16x16 tile at cols [16w,16w+16).
// A fragment (16x4 f32): lane<16 -> K pair {k,k+1}, lane>=16 -> {k+2,k+3}; row = lane&15.
// B fragment (4x16 f32): mirrored layout, col = lane&15.
// C/D (16x16 f32, 8 VGPRs): m = v + 8*(lane>=16), n = lane&15.
// Loads are unconditional (row clamped to N-1, a no-op when N % 16 == 0);
// the store path uses a single uniform bounds test so the hot loop has no
// exec-mask manipulation around the WMMA.
__global__ void gemm128(const float* __restrict__ A, int K, int N,
                        const float* __restrict__ W,
                        float* __restrict__ Out,
                        const float* __restrict__ bias, int do_relu) {
  const int lane = threadIdx.x & 31;
  const int wave = threadIdx.x >> 5;
  const int m0 = blockIdx.x * 16;
  const int n0 = wave * 16;
  const int half = lane >> 4;
  const int r = lane & 15;
  const int n = n0 + r;

  int row = m0 + r;
  row = (row < N) ? row : (N - 1);  // branchless clamp, keeps loads in-bounds

  const float* aBase = A + (size_t)row * K + half * 2;          // 8B aligned
  const float* wBase = W + (size_t)(half * 2) * 128 + n;

  v8f acc = {};
#pragma unroll 4
  for (int k = 0; k < K; k += 4) {
    v2f a = *(const v2f*)(aBase + k);          // global_load_b64
    v2f b;
    b.x = wBase[(size_t)k * 128];              // row ka of W
    b.y = wBase[(size_t)k * 128 + 128];        // row ka+1 of W
    acc = __builtin_amdgcn_wmma_f32_16x16x4_f32(false, a, false, b, (short)0, acc,
                                                false, false);
  }

  const float badd = bias ? bias[n] : 0.0f;
  float* outBase = Out + (size_t)(m0 + half * 8) * 128 + n;
  if (m0 + 16 <= N) {            // uniform fast path: full tile in bounds
#pragma unroll
    for (int v = 0; v < 8; ++v) {
      float val = acc[v] + badd;
      if (do_relu) val = fmaxf(val, 0.0f);
      outBase[(size_t)v * 128] = val;
    }
  } else {
#pragma unroll
    for (int v = 0; v < 8; ++v) {
      int m = m0 + v + half * 8;
      if (m < N) {
        float val = acc[v] + badd;
        if (do_relu) val = fmaxf(val, 0.0f);
        Out[(size_t)m * 128 + n] = val;
      }
    }
  }
}

// ---------------- edge scatter: one wave per edge, float4 per lane ----------------
__global__ void scatter_edges(const float* __restrict__ hw,
                              const int* __restrict__ src, const int* __restrict__ dst,
                              const float* __restrict__ dis,
                              float* __restrict__ agg, int E) {
  int e = (int)((blockIdx.x * (size_t)blockDim.x + threadIdx.x) >> 5);
  int lane = threadIdx.x & 31;
  if (e >= E) return;
  int s = src[e], d = dst[e];
  float w = dis[s] * dis[d];
  const float4* rowp = (const float4*)(hw + (size_t)s * 128);
  float4 v = rowp[lane];
  float* out = agg + (size_t)d * 128 + lane * 4;
  unsafeAtomicAdd(out + 0, v.x * w);
  unsafeAtomicAdd(out + 1, v.y * w);
  unsafeAtomicAdd(out + 2, v.z * w);
  unsafeAtomicAdd(out + 3, v.w * w);
}

// ---------------- self-loop + bias (+emb tap) + relu ----------------
__global__ void finalize(const float* __restrict__ agg, const float* __restrict__ hw,
                         const float* __restrict__ dis, const float* __restrict__ bias,
                         float* __restrict__ h_out, float* __restrict__ emb, int N) {
  int idx = blockIdx.x * blockDim.x + threadIdx.x;
  if (idx >= N * 128) return;
  int i = idx >> 7, f = idx & 127;
  float di = dis[i];
  float val = agg[idx] + hw[idx] * di * di + bias[f];
  if (emb) emb[idx] = val;
  h_out[idx] = fmaxf(val, 0.0f);
}

// ---------------- head: pred[N,3] = t[N,128] @ Wr2[128,3] + br2 ----------------
__global__ void pred_kernel(const float* __restrict__ t, const float* __restrict__ Wr2,
                            const float* __restrict__ br2, float* __restrict__ pred, int N) {
  int i = blockIdx.x * blockDim.x + threadIdx.x;
  if (i >= N) return;
  float a0 = br2[0], a1 = br2[1], a2 = br2[2];
  const float* row = t + (size_t)i * 128;
#pragma unroll 4
  for (int k = 0; k < 128; ++k) {
    float tv = row[k];
    a0 += tv * Wr2[k * 3 + 0];
    a1 += tv * Wr2[k * 3 + 1];
    a2 += tv * Wr2[k * 3 + 2];
  }
  pred[(size_t)i * 3 + 0] = a0;
  pred[(size_t)i * 3 + 1] = a1;
  pred[(size_t)i * 3 + 2] = a2;
}

extern "C" void kernel_launch(void* const* d_in, const int* in_sizes, int n_in,
                              void* d_out, int out_size, void* d_ws, size_t ws_size,
                              hipStream_t stream) {
  const float* x   = (const float*)d_in[0];
  const float* xm  = (const float*)d_in[1];
  const int*   ei  = (const int*)d_in[2];
  const float* W0  = (const float*)d_in[3];
  const float* b0  = (const float*)d_in[4];
  const float* Wh  = (const float*)d_in[5];
  const float* bh  = (const float*)d_in[6];
  const float* Wr1 = (const float*)d_in[7];
  const float* br1 = (const float*)d_in[8];
  const float* Wr2 = (const float*)d_in[9];
  const float* br2 = (const float*)d_in[10];

  const int N = in_sizes[0] / 10;
  const int E = in_sizes[2] / 2;
  const int* srcv = ei;
  const int* dstv = ei + E;

  float* bufA = (float*)d_ws;                 // h (layer input; N x 128, h0 uses N x 16)
  float* bufB = bufA + (size_t)N * 128;       // hw = h @ W
  float* bufC = bufB + (size_t)N * 128;       // agg
  float* dis  = bufC + (size_t)N * 128;       // deg -> deg_inv_sqrt

  float* emb_out  = (float*)d_out;
  float* pred_out = emb_out + (size_t)N * 128;

  const int mblocks  = (N + 15) / 16;
  const int nfblocks = (int)(((size_t)N * 128 + 255) / 256);
  const int sblocks  = (int)(((size_t)E * 32 + 255) / 256);

  // degrees & normalization
  hipMemsetAsync(dis, 0, (size_t)N * sizeof(float), stream);
  deg_kernel<<<(E + 255) / 256, 256, 0, stream>>>(dstv, dis, E);
  rsqrt_kernel<<<(N + 255) / 256, 256, 0, stream>>>(dis, N);

  // input features
  build_h0<<<(int)(((size_t)N * 16 + 255) / 256), 256, 0, stream>>>(x, xm, bufA, N);

  // layer 0: 16 -> 128
  gemm128<<<mblocks, 256, 0, stream>>>(bufA, 16, N, W0, bufB, nullptr, 0);
  hipMemsetAsync(bufC, 0, (size_t)N * 128 * sizeof(float), stream);
  scatter_edges<<<sblocks, 256, 0, stream>>>(bufB, srcv, dstv, dis, bufC, E);
  finalize<<<nfblocks, 256, 0, stream>>>(bufC, bufB, dis, b0, bufA, nullptr, N);

  // hidden layers: 128 -> 128 (emb tapped from the last one, pre-ReLU)
  for (int l = 0; l < 4; ++l) {
    gemm128<<<mblocks, 256, 0, stream>>>(bufA, 128, N, Wh + (size_t)l * 128 * 128,
                                         bufB, nullptr, 0);
    hipMemsetAsync(bufC, 0, (size_t)N * 128 * sizeof(float), stream);
    scatter_edges<<<sblocks, 256, 0, stream>>>(bufB, srcv, dstv, dis, bufC, E);
    finalize<<<nfblocks, 256, 0, stream>>>(bufC, bufB, dis, bh + (size_t)l * 128, bufA,
                                           (l == 3) ? emb_out : nullptr, N);
  }

  // head: relu(h @ Wr1 + br1) @ Wr2 + br2
  gemm128<<<mblocks, 256, 0, stream>>>(bufA, 128, N, Wr1, bufB, br1, 1);
  pred_kernel<<<(N + 255) / 256, 256, 0, stream>>>(bufB, Wr2, br2, pred_out, N);
}